// NGCF_3693671874623
// MI455X (gfx1250) — compile-verified
//
#include <hip/hip_runtime.h>

typedef __attribute__((ext_vector_type(2))) float v2f;
typedef __attribute__((ext_vector_type(8))) float v8f;

#define EMB 64
#define OUTW 256

// ---------------- zero workspace ----------------
__global__ void zero_kernel(float* __restrict__ p, int n4) {
    int i = blockIdx.x * blockDim.x + threadIdx.x;
    if (i < n4) ((float4*)p)[i] = make_float4(0.f, 0.f, 0.f, 0.f);
}

// ---------------- out[:,0:64] = emb_table ----------------
__global__ void copy_emb_kernel(const float* __restrict__ emb, float* __restrict__ out, int n) {
    int i = blockIdx.x * blockDim.x + threadIdx.x;   // n*16 float4 groups
    if (i >= n * 16) return;
    int node = i >> 4, c4 = i & 15;
    float4 v = ((const float4*)(emb + (long long)node * EMB))[c4];
    ((float4*)(out + (long long)node * OUTW))[c4] = v;
}

// ---------------- SpMM: y[rows[e]] += vals[e] * x[cols[e]] ----------------
// 16 threads per edge, float4 per thread -> coalesced gathers, HW fp32 atomics.
__global__ void spmm_kernel(const float* __restrict__ vals, const int* __restrict__ rows,
                            const int* __restrict__ cols, const float* __restrict__ x,
                            float* __restrict__ y, int n_edges) {
    long long t = (long long)blockIdx.x * blockDim.x + threadIdx.x;
    int e = (int)(t >> 4);
    if (e >= n_edges) return;
    int c = (int)(t & 15) << 2;
    float v = vals[e];
    float4 xv = *(const float4*)(x + (long long)cols[e] * EMB + c);
    float* yp = y + (long long)rows[e] * EMB + c;
    unsafeAtomicAdd(yp + 0, v * xv.x);
    unsafeAtomicAdd(yp + 1, v * xv.y);
    unsafeAtomicAdd(yp + 2, v * xv.z);
    unsafeAtomicAdd(yp + 3, v * xv.w);
}

// ---------------- dense: node_out = leaky(agg@W1^T+b1) + leaky((agg*node)@W2^T+b2) ----
// One wave per 16-row tile, full-precision f32 WMMA 16x16x4.
__global__ __launch_bounds__(128) void dense_kernel(
        const float* __restrict__ agg, const float* __restrict__ node_in,
        const float* __restrict__ W1, const float* __restrict__ b1,
        const float* __restrict__ W2, const float* __restrict__ b2,
        float* __restrict__ node_out, int n_rows) {
    int wave = threadIdx.x >> 5;
    int lane = threadIdx.x & 31;
    int tile = blockIdx.x * 4 + wave;
    int row0 = tile * 16;
    if (row0 >= n_rows) return;           // wave-uniform: EXEC stays all-ones for WMMA

    int half = lane >> 4;                 // 0: K={4k,4k+1}, 1: K={4k+2,4k+3}
    int l    = lane & 15;
    int arow = row0 + l;
    int koff = half * 2;

    // A fragments for agg and agg*node, all 16 K-steps (K=64)
    v2f a_agg[16], a_mul[16];
#pragma unroll
    for (int kk = 0; kk < 16; ++kk) {
        v2f ag = *(const v2f*)(agg     + (long long)arow * EMB + kk * 4 + koff);
        v2f nd = *(const v2f*)(node_in + (long long)arow * EMB + kk * 4 + koff);
        a_agg[kk] = ag;
        a_mul[kk] = ag * nd;
    }

#pragma unroll
    for (int j = 0; j < 4; ++j) {         // 4 column tiles of 16
        float bb1 = b1[j * 16 + l];
        float bb2 = b2[j * 16 + l];
        v8f acc1, acc2;
#pragma unroll
        for (int i = 0; i < 8; ++i) { acc1[i] = bb1; acc2[i] = bb2; }

        // B = W^T: B[k][n] = W[n][k] -> contiguous float2 from row-major W
        const float* w1p = W1 + (long long)(j * 16 + l) * EMB + koff;
        const float* w2p = W2 + (long long)(j * 16 + l) * EMB + koff;
#pragma unroll
        for (int kk = 0; kk < 16; ++kk) {
            v2f bw1 = *(const v2f*)(w1p + kk * 4);
            v2f bw2 = *(const v2f*)(w2p + kk * 4);
            acc1 = __builtin_amdgcn_wmma_f32_16x16x4_f32(
                false, a_agg[kk], false, bw1, (short)0, acc1, false, false);
            acc2 = __builtin_amdgcn_wmma_f32_16x16x4_f32(
                false, a_mul[kk], false, bw2, (short)0, acc2, false, false);
        }

        // D layout: VGPR i -> row (half*8 + i), lane -> column (j*16 + l)
#pragma unroll
        for (int i = 0; i < 8; ++i) {
            float x1 = acc1[i]; x1 = (x1 >= 0.f) ? x1 : 0.2f * x1;
            float x2 = acc2[i]; x2 = (x2 >= 0.f) ? x2 : 0.2f * x2;
            int r = row0 + half * 8 + i;
            node_out[(long long)r * EMB + j * 16 + l] = x1 + x2;
        }
    }
}

// ---------------- out[:, col_base:col_base+64] = l2norm(node) ----------------
// One wave32 per row, shuffle reduction.
__global__ void l2norm_kernel(const float* __restrict__ node, float* __restrict__ out,
                              int n, int col_base) {
    int wave = threadIdx.x >> 5;
    int lane = threadIdx.x & 31;
    int row = blockIdx.x * (blockDim.x >> 5) + wave;
    if (row >= n) return;
    v2f v = *(const v2f*)(node + (long long)row * EMB + lane * 2);
    float ss = v.x * v.x + v.y * v.y;
#pragma unroll
    for (int m = 16; m >= 1; m >>= 1) ss += __shfl_xor(ss, m, 32);
    float inv = 1.f / fmaxf(sqrtf(ss), 1e-12f);
    float* op = out + (long long)row * OUTW + col_base + lane * 2;
    op[0] = v.x * inv;
    op[1] = v.y * inv;
}

extern "C" void kernel_launch(void* const* d_in, const int* in_sizes, int n_in,
                              void* d_out, int out_size, void* d_ws, size_t ws_size,
                              hipStream_t stream) {
    const float* emb  = (const float*)d_in[0];
    const float* vals = (const float*)d_in[1];
    const int*   rows = (const int*)d_in[2];
    const int*   cols = (const int*)d_in[3];
    int N = in_sizes[0] / EMB;
    int E = in_sizes[1];
    size_t NE = (size_t)N * EMB;

    float* agg   = (float*)d_ws;          // 3 x N*64 f32 buffers in workspace
    float* nodeA = agg + NE;
    float* nodeB = nodeA + NE;
    float* out   = (float*)d_out;

    copy_emb_kernel<<<(N * 16 + 255) / 256, 256, 0, stream>>>(emb, out, N);

    const float* node_in = emb;
    float* node_bufs[2] = { nodeA, nodeB };
    for (int layer = 0; layer < 3; ++layer) {
        const float* W1 = (const float*)d_in[4 + layer * 4 + 0];
        const float* b1 = (const float*)d_in[4 + layer * 4 + 1];
        const float* W2 = (const float*)d_in[4 + layer * 4 + 2];
        const float* b2 = (const float*)d_in[4 + layer * 4 + 3];
        float* node_out = node_bufs[layer & 1];

        int n4 = (int)(NE / 4);
        zero_kernel<<<(n4 + 255) / 256, 256, 0, stream>>>(agg, n4);

        long long sp_threads = (long long)E * 16;
        spmm_kernel<<<(int)((sp_threads + 255) / 256), 256, 0, stream>>>(
            vals, rows, cols, node_in, agg, E);

        int tiles = (N + 15) / 16;
        dense_kernel<<<(tiles + 3) / 4, 128, 0, stream>>>(
            agg, node_in, W1, b1, W2, b2, node_out, N);

        l2norm_kernel<<<(N + 7) / 8, 256, 0, stream>>>(
            node_out, out, N, (layer + 1) * EMB);

        node_in = node_out;
    }
}